// ESNet_10763188043993
// MI455X (gfx1250) — compile-verified
//
#include <hip/hip_runtime.h>
#include <hip/hip_bf16.h>
#include <math.h>

typedef __attribute__((ext_vector_type(2))) float v2f;
typedef __attribute__((ext_vector_type(8))) float v8f;

#define BATCH 4096
#define NCH   8
#define TLEN  128
#define F1C   16
#define F2C   32
#define TP    60
#define HID   32
#define GATES 128
#define FEAT  3840
#define NCLS  12

// ------------------------------------------------------------------ utility
__global__ void zero_kernel(float* p, int n) {
    int i = blockIdx.x * blockDim.x + threadIdx.x;
    if (i < n) p[i] = 0.f;
}

// cos table: tab[k*60+t] = cos(2*pi*((k*t)%60)/60)  (exact angle reduction)
__global__ void costab_kernel(float* tab) {
    int i = blockIdx.x * blockDim.x + threadIdx.x;
    if (i < TP * TP) {
        int k = i / TP, t = i % TP;
        int m = (k * t) % TP;
        tab[i] = cosf(6.2831853071795864769f * (float)m / (float)TP);
    }
}

// ------------------------------------------------------------------ conv1
// x: [B][1][8][128]  w: [16][8]  -> y1: [B][16][128]
__global__ __launch_bounds__(256)
void conv1_kernel(const float* __restrict__ x, const float* __restrict__ w,
                  const float* __restrict__ bias, float* __restrict__ y1) {
    __shared__ float s_w[F1C * NCH];
    int tid = threadIdx.x;
    if (tid < F1C * NCH) s_w[tid] = w[tid];
    __syncthreads();
    int gid = blockIdx.x * 256 + tid;            // over B*T = 524288
    int b = gid >> 7, t = gid & 127;
    float xv[NCH];
#pragma unroll
    for (int ch = 0; ch < NCH; ++ch) xv[ch] = x[(size_t)b * (NCH * TLEN) + ch * TLEN + t];
#pragma unroll
    for (int f = 0; f < F1C; ++f) {
        float a = bias[f];
#pragma unroll
        for (int ch = 0; ch < NCH; ++ch) a += xv[ch] * s_w[f * NCH + ch];
        y1[(size_t)b * (F1C * TLEN) + f * TLEN + t] = a;
    }
}

// ------------------------------------------------- per-channel sum / sumsq
// src layout [nouter][nchan][len]; stats[ch]=sum, stats[nchan+ch]=sumsq
__global__ __launch_bounds__(256)
void stats_kernel(const float* __restrict__ src, int nouter, int nchan, int len,
                  float* __restrict__ stats) {
    int ch = blockIdx.y;
    float s = 0.f, q = 0.f;
    size_t total = (size_t)nouter * len;
    for (size_t i = blockIdx.x * 256 + threadIdx.x; i < total; i += (size_t)gridDim.x * 256) {
        size_t b = i / (size_t)len, l = i % (size_t)len;
        float v = src[(b * nchan + ch) * len + l];
        s += v; q += v * v;
    }
    __shared__ float rs[256], rq[256];
    rs[threadIdx.x] = s; rq[threadIdx.x] = q;
    __syncthreads();
    for (int off = 128; off > 0; off >>= 1) {
        if ((int)threadIdx.x < off) { rs[threadIdx.x] += rs[threadIdx.x + off]; rq[threadIdx.x] += rq[threadIdx.x + off]; }
        __syncthreads();
    }
    if (threadIdx.x == 0) { atomicAdd(&stats[ch], rs[0]); atomicAdd(&stats[nchan + ch], rq[0]); }
}

__global__ void finalize_bn(const float* __restrict__ stats, const float* __restrict__ g,
                            const float* __restrict__ b, int nchan, float invN,
                            float* __restrict__ params) {
    int c = blockIdx.x * blockDim.x + threadIdx.x;
    if (c < nchan) {
        float m = stats[c] * invN;
        float v = stats[nchan + c] * invN - m * m;
        float sc = g[c] * rsqrtf(v + 1e-5f);
        params[c] = sc;
        params[nchan + c] = b[c] - m * sc;
    }
}

// in-place bn (affine) + prelu on [nouter][nchan][len]
__global__ __launch_bounds__(256)
void bnprelu_kernel(float* __restrict__ data, const float* __restrict__ params,
                    int nchan, int len, size_t total, const float* __restrict__ alpha) {
    size_t i = (size_t)blockIdx.x * 256 + threadIdx.x;
    if (i < total) {
        int ch = (int)((i / (size_t)len) % (size_t)nchan);
        float v = data[i] * params[ch] + params[nchan + ch];
        data[i] = (v >= 0.f) ? v : alpha[0] * v;
    }
}

// ------------------------------------------------------------------ conv2
// y1: [B][16][128], w2: [32][16][10], stride 2 -> y2: [B][32][60]
__global__ __launch_bounds__(64)
void conv2_kernel(const float* __restrict__ y1, const float* __restrict__ w2,
                  const float* __restrict__ b2, float* __restrict__ y2) {
    __shared__ float s_in[F1C * TLEN];     // 8 KB
    __shared__ float s_w[F2C * F1C * 10];  // 20 KB
    int b = blockIdx.x, tid = threadIdx.x;
    for (int i = tid; i < F1C * TLEN; i += 64) s_in[i] = y1[(size_t)b * (F1C * TLEN) + i];
    for (int i = tid; i < F2C * F1C * 10; i += 64) s_w[i] = w2[i];
    __syncthreads();
    if (tid < TP) {
        for (int f2 = 0; f2 < F2C; ++f2) {
            float a = b2[f2];
            for (int f1 = 0; f1 < F1C; ++f1) {
                const float* in = &s_in[f1 * TLEN + 2 * tid];
                const float* wk = &s_w[(f2 * F1C + f1) * 10];
#pragma unroll
                for (int k = 0; k < 10; ++k) a += in[k] * wk[k];
            }
            y2[(size_t)b * (F2C * TP) + f2 * TP + tid] = a;
        }
    }
}

// ------------------------------------------------------------------ fft (real part of DFT, len 60)
__global__ __launch_bounds__(64)
void fft_kernel(const float* __restrict__ src, const float* __restrict__ tab,
                float* __restrict__ dst) {
    __shared__ float row[TP];
    size_t rc = blockIdx.x;                 // over B*F2C
    int tid = threadIdx.x;
    if (tid < TP) row[tid] = src[rc * TP + tid];
    __syncthreads();
    if (tid < TP) {
        float a = 0.f;
        const float* tb = tab + tid * TP;
#pragma unroll 6
        for (int t = 0; t < TP; ++t) a += row[t] * tb[t];
        dst[rc * TP + tid] = a;
    }
}

// ------------------------------------------------------------------ LSTM (WMMA f32 16x16x4)
// xin viewed as [60][4096][32]; rout viewed as [60][4096][64]
// each block: 16 batch rows, 8 waves -> wave w owns gate cols [16w,16w+16)
__global__ __launch_bounds__(256)
void lstm_kernel(const float* __restrict__ xin,
                 const float* __restrict__ wih, const float* __restrict__ whh,
                 const float* __restrict__ bih, const float* __restrict__ bhh,
                 float* __restrict__ rout, int backward, int outoff, int accumulate) {
    __shared__ float s_x[16][HID];
    __shared__ float s_h[16][HID];
    __shared__ float s_c[16][HID];
    __shared__ float s_g[16][GATES];

    const int tid = threadIdx.x;
    const int lane = tid & 31;
    const int wave = tid >> 5;
    const int row0 = blockIdx.x * 16;
    const int n0 = wave * 16;

    for (int i = tid; i < 16 * HID; i += 256) { (&s_h[0][0])[i] = 0.f; (&s_c[0][0])[i] = 0.f; }

    // B fragment layout (4x16 f32, 2 VGPRs): elem(K = v + 2*(lane/16), N = lane%16)
    const int ncol = n0 + (lane & 15);
    const int khalf = (lane >> 4) * 2;
    v2f bw_ih[8], bw_hh[8];
#pragma unroll
    for (int kc = 0; kc < 8; ++kc) {
        int k0 = kc * 4 + khalf;
        bw_ih[kc].x = wih[ncol * HID + k0];
        bw_ih[kc].y = wih[ncol * HID + k0 + 1];
        bw_hh[kc].x = whh[ncol * HID + k0];
        bw_hh[kc].y = whh[ncol * HID + k0 + 1];
    }
    const float bias = bih[ncol] + bhh[ncol];
    const int am = lane & 15;       // A row (16x4 f32: M = lane%16, K = v + 2*(lane/16))
    __syncthreads();

    for (int step = 0; step < TP; ++step) {
        const int t = backward ? (TP - 1 - step) : step;
        const float* xsrc = xin + (size_t)t * BATCH * HID + (size_t)row0 * HID;
        for (int i = tid; i < 16 * HID; i += 256) (&s_x[0][0])[i] = xsrc[i];
        __syncthreads();

        v8f acc;
#pragma unroll
        for (int v = 0; v < 8; ++v) acc[v] = bias;
#pragma unroll
        for (int kc = 0; kc < 8; ++kc) {
            int k0 = kc * 4 + khalf;
            v2f ax, ah;
            ax.x = s_x[am][k0]; ax.y = s_x[am][k0 + 1];
            ah.x = s_h[am][k0]; ah.y = s_h[am][k0 + 1];
            acc = __builtin_amdgcn_wmma_f32_16x16x4_f32(false, ax, false, bw_ih[kc],
                                                        (short)0, acc, false, false);
            acc = __builtin_amdgcn_wmma_f32_16x16x4_f32(false, ah, false, bw_hh[kc],
                                                        (short)0, acc, false, false);
        }
        // C layout: elem(M = v + 8*(lane/16), N = lane%16)
        {
            int mb = (lane >> 4) * 8;
#pragma unroll
            for (int v = 0; v < 8; ++v) s_g[mb + v][n0 + (lane & 15)] = acc[v];
        }
        __syncthreads();

        for (int i = tid; i < 16 * HID; i += 256) {
            int r = i >> 5, hd = i & 31;
            float gi = s_g[r][hd];
            float gf = s_g[r][HID + hd];
            float gg = s_g[r][2 * HID + hd];
            float go = s_g[r][3 * HID + hd];
            float si = 1.f / (1.f + expf(-gi));
            float sf = 1.f / (1.f + expf(-gf));
            float so = 1.f / (1.f + expf(-go));
            float c = sf * s_c[r][hd] + si * tanhf(gg);
            float h = so * tanhf(c);
            s_c[r][hd] = c;
            s_h[r][hd] = h;
            size_t oidx = (size_t)t * BATCH * 64 + (size_t)(row0 + r) * 64 + outoff + hd;
            if (accumulate) rout[oidx] += h; else rout[oidx] = h;
        }
        __syncthreads();
    }
}

// ------------------------------------------------------------------ bnf (per-feature over 4096 rows)
__global__ __launch_bounds__(256)
void bnf_kernel(const float* __restrict__ r, const float* __restrict__ g,
                const float* __restrict__ b, float* __restrict__ params) {
    int j = blockIdx.x * 256 + threadIdx.x;
    if (j >= FEAT) return;
    float s = 0.f, q = 0.f;
    for (int i = 0; i < BATCH; ++i) {
        float v = r[(size_t)i * FEAT + j];
        s += v; q += v * v;
    }
    float m = s * (1.f / BATCH);
    float var = q * (1.f / BATCH) - m * m;
    float sc = g[j] * rsqrtf(var + 1e-5f);
    params[j] = sc;
    params[FEAT + j] = b[j] - m * sc;
}

__global__ __launch_bounds__(256)
void bnf_apply(float* __restrict__ data, const float* __restrict__ params) {
    size_t i = (size_t)blockIdx.x * 256 + threadIdx.x;
    if (i < (size_t)BATCH * FEAT) {
        int j = (int)(i % FEAT);
        data[i] = data[i] * params[j] + params[FEAT + j];
    }
}

// ------------------------------------------------------------------ corr + final conv
// out[0 .. B*12)   = conv result, out[B*12 .. 2*B*12) = corr
__global__ __launch_bounds__(96)
void corr_kernel(const float* __restrict__ sx, const float* __restrict__ st,
                 const float* __restrict__ wf, const float* __restrict__ bf,
                 float* __restrict__ out) {
    int b = blockIdx.x, tid = threadIdx.x;
    int c = tid % NCLS, s = tid / NCLS;       // 12 classes x 8 partials
    __shared__ float rxt[8][NCLS], rxx[8][NCLS], rtt[8][NCLS];
    __shared__ float corr[NCLS];
    float axt = 0.f, axx = 0.f, att = 0.f;
    const float* px = sx + (size_t)b * FEAT;
    const float* pt = st + (size_t)b * FEAT;
    for (int p = s; p < 320; p += 8) {
        float xv = px[p * NCLS + c], tv = pt[p * NCLS + c];
        axt += xv * tv; axx += xv * xv; att += tv * tv;
    }
    rxt[s][c] = axt; rxx[s][c] = axx; rtt[s][c] = att;
    __syncthreads();
    if (s == 0) {
        float xt = 0.f, xx = 0.f, tt = 0.f;
        for (int k = 0; k < 8; ++k) { xt += rxt[k][c]; xx += rxx[k][c]; tt += rtt[k][c]; }
        float cr = xt / sqrtf(tt) / sqrtf(xx);
        corr[c] = cr;
        out[(size_t)BATCH * NCLS + (size_t)b * NCLS + c] = cr;
    }
    __syncthreads();
    if (s == 0) {
        float a = bf[0];
#pragma unroll
        for (int k = 0; k < 5; ++k) {
            int idx = c - 2 + k;
            if (idx >= 0 && idx < NCLS) a += corr[idx] * wf[k];
        }
        out[(size_t)b * NCLS + c] = a;
    }
}

// ------------------------------------------------------------------ launch
extern "C" void kernel_launch(void* const* d_in, const int* in_sizes, int n_in,
                              void* d_out, int out_size, void* d_ws, size_t ws_size,
                              hipStream_t stream) {
    const float* x        = (const float*)d_in[0];
    const float* templ    = (const float*)d_in[1];
    const float* conv1_w  = (const float*)d_in[2];
    const float* conv1_b  = (const float*)d_in[3];
    const float* bn1_g    = (const float*)d_in[4];
    const float* bn1_b    = (const float*)d_in[5];
    const float* prelu1   = (const float*)d_in[6];
    const float* conv2_w  = (const float*)d_in[7];
    const float* conv2_b  = (const float*)d_in[8];
    const float* bn2_g    = (const float*)d_in[9];
    const float* bn2_b    = (const float*)d_in[10];
    const float* prelu2   = (const float*)d_in[11];
    const float* wih_f    = (const float*)d_in[12];
    const float* whh_f    = (const float*)d_in[13];
    const float* bih_f    = (const float*)d_in[14];
    const float* bhh_f    = (const float*)d_in[15];
    const float* wih_r    = (const float*)d_in[16];
    const float* whh_r    = (const float*)d_in[17];
    const float* bih_r    = (const float*)d_in[18];
    const float* bhh_r    = (const float*)d_in[19];
    const float* bnf_g    = (const float*)d_in[20];
    const float* bnf_b    = (const float*)d_in[21];
    const float* convf_w  = (const float*)d_in[22];
    const float* convf_b  = (const float*)d_in[23];
    float* outp = (float*)d_out;

    float* W = (float*)d_ws;
    float* y1      = W;                       // 8,388,608
    float* osig    = y1   + 8388608;          // 7,864,320
    float* fftm    = osig + 7864320;          // 7,864,320
    float* r_x     = fftm + 7864320;          // 15,728,640
    float* r_t     = r_x  + 15728640;         // 15,728,640
    float* sreg    = r_t  + 15728640;
    float* stats1  = sreg;                    // 32
    float* stats2  = sreg + 32;               // 64
    float* params1 = sreg + 128;              // 32
    float* params2 = sreg + 192;              // 64
    float* paramsF = sreg + 256;              // 7680
    float* costab  = sreg + 8192;             // 3600

    costab_kernel<<<(TP * TP + 255) / 256, 256, 0, stream>>>(costab);

    for (int branch = 0; branch < 2; ++branch) {
        const float* inp = branch ? templ : x;
        float* rbuf = branch ? r_t : r_x;

        zero_kernel<<<1, 128, 0, stream>>>(sreg, 96);
        conv1_kernel<<<(BATCH * TLEN) / 256, 256, 0, stream>>>(inp, conv1_w, conv1_b, y1);
        stats_kernel<<<dim3(64, F1C), 256, 0, stream>>>(y1, BATCH, F1C, TLEN, stats1);
        finalize_bn<<<1, 128, 0, stream>>>(stats1, bn1_g, bn1_b, F1C,
                                           1.f / (float)(BATCH * TLEN), params1);
        bnprelu_kernel<<<(8388608 + 255) / 256, 256, 0, stream>>>(
            y1, params1, F1C, TLEN, (size_t)8388608, prelu1);

        conv2_kernel<<<BATCH, 64, 0, stream>>>(y1, conv2_w, conv2_b, osig);
        stats_kernel<<<dim3(32, F2C), 256, 0, stream>>>(osig, BATCH, F2C, TP, stats2);
        finalize_bn<<<1, 128, 0, stream>>>(stats2, bn2_g, bn2_b, F2C,
                                           1.f / (float)(BATCH * TP), params2);
        bnprelu_kernel<<<(7864320 + 255) / 256, 256, 0, stream>>>(
            osig, params2, F2C, TP, (size_t)7864320, prelu2);

        fft_kernel<<<BATCH * F2C, 64, 0, stream>>>(osig, costab, fftm);

        // r1 (raw signal) write, then r2 (fft) accumulate; fwd -> cols 0..31, bwd -> 32..63
        lstm_kernel<<<BATCH / 16, 256, 0, stream>>>(osig, wih_f, whh_f, bih_f, bhh_f, rbuf, 0,  0, 0);
        lstm_kernel<<<BATCH / 16, 256, 0, stream>>>(osig, wih_r, whh_r, bih_r, bhh_r, rbuf, 1, 32, 0);
        lstm_kernel<<<BATCH / 16, 256, 0, stream>>>(fftm, wih_f, whh_f, bih_f, bhh_f, rbuf, 0,  0, 1);
        lstm_kernel<<<BATCH / 16, 256, 0, stream>>>(fftm, wih_r, whh_r, bih_r, bhh_r, rbuf, 1, 32, 1);

        bnf_kernel<<<(FEAT + 255) / 256, 256, 0, stream>>>(rbuf, bnf_g, bnf_b, paramsF);
        bnf_apply<<<((size_t)BATCH * FEAT + 255) / 256, 256, 0, stream>>>(rbuf, paramsF);
    }

    corr_kernel<<<BATCH, 96, 0, stream>>>(r_x, r_t, convf_w, convf_b, outp);
    (void)in_sizes; (void)n_in; (void)out_size; (void)ws_size;
}